// CORALClassifier_15522011808336
// MI455X (gfx1250) — compile-verified
//
#include <hip/hip_runtime.h>
#include <hip/hip_bf16.h>

// CDNA5 / gfx1250: wave32, WMMA fp32 16x16x4.

typedef __attribute__((ext_vector_type(2))) float v2f;
typedef __attribute__((ext_vector_type(8))) float v8f;

#define GCN_N_NODES 100000
#define GCN_N_EDGES 1600000
#define GCN_FEAT 128
#define GCN_KOUT 4  // NUM_CLASSES - 1

// ---------------- utility ----------------

__global__ void gcn_zero_f32(float* __restrict__ p, int n) {
    int i = blockIdx.x * blockDim.x + threadIdx.x;
    if (i < n) p[i] = 0.0f;
}

// ---------------- degree / norm ----------------

__global__ void gcn_deg_count(const int* __restrict__ dst, float* __restrict__ deg, int e) {
    int i = blockIdx.x * blockDim.x + threadIdx.x;
    if (i < e) atomicAdd(&deg[dst[i]], 1.0f);
}

__global__ void gcn_deg_finalize(float* __restrict__ deg, int n) {
    int i = blockIdx.x * blockDim.x + threadIdx.x;
    if (i < n) deg[i] = rsqrtf(deg[i] + 1.0f);  // deg includes self-loop (+1)
}

// ---------------- GEMM: H[M,128] = X[M,128] @ W[128,128] via V_WMMA_F32_16X16X4_F32 ----
// One wave computes a full 16x128 output strip (8 accumulator tiles, 64 VGPRs),
// so the A fragment (16x4, 2 VGPRs) is loaded once per K-step and reused by all
// 8 WMMAs -> 8x less A traffic, back-to-back wmma issue.
//
// A (16x4 f32) lane layout: lanes 0-15 rows M=0..15, VGPR0=K0,VGPR1=K1; lanes 16-31 K2/K3.
// B (4x16 f32): lanes 0-15 cols N=0..15 with K0/K1; lanes 16-31 same cols K2/K3.
// C/D (16x16 f32): VGPR i -> row m0+i (lanes 0-15) / m0+8+i (lanes 16-31), col = lane&15.

__global__ void gcn_gemm128_wmma(const float* __restrict__ X,
                                 const float* __restrict__ W,
                                 float* __restrict__ H, int M) {
    const int waves_per_blk = blockDim.x >> 5;
    const int tile_m = blockIdx.x * waves_per_blk + (threadIdx.x >> 5);
    if (tile_m * 16 >= M) return;  // wave-uniform: EXEC stays all-ones for WMMA

    const int lane = threadIdx.x & 31;
    const int hi = lane >> 4;     // 0: lanes 0-15, 1: lanes 16-31
    const int r = lane & 15;
    const int m0 = tile_m * 16;

    const float* __restrict__ xrow = X + (size_t)(m0 + r) * GCN_FEAT;
    const int koff = hi ? 2 : 0;

    v8f acc[8];
#pragma unroll
    for (int t = 0; t < 8; ++t) acc[t] = (v8f){};

    for (int k = 0; k < GCN_FEAT; k += 4) {
        const int ka = k + koff;
        v2f a;
        a.x = xrow[ka];
        a.y = xrow[ka + 1];
        const float* __restrict__ w0 = W + (size_t)ka * GCN_FEAT + r;
        const float* __restrict__ w1 = w0 + GCN_FEAT;
#pragma unroll
        for (int t = 0; t < 8; ++t) {
            v2f b;
            b.x = w0[t * 16];
            b.y = w1[t * 16];
            // 8 args: (neg_a, A, neg_b, B, c_mod, C, reuse_a, reuse_b)
            acc[t] = __builtin_amdgcn_wmma_f32_16x16x4_f32(
                false, a, false, b, (short)0, acc[t], false, false);
        }
    }

    const int mbase = m0 + (hi ? 8 : 0);
#pragma unroll
    for (int t = 0; t < 8; ++t) {
        float* __restrict__ hcol = H + (size_t)mbase * GCN_FEAT + t * 16 + r;
#pragma unroll
        for (int i = 0; i < 8; ++i)
            hcol[(size_t)i * GCN_FEAT] = acc[t][i];
    }
}

// ---------------- edge scatter: agg[dst] += h[src] * dis[src]*dis[dst] ----------------
// 32 lanes per edge, one float4 feature chunk per lane (coalesced 512B gather).

__global__ void gcn_scatter(const int* __restrict__ src, const int* __restrict__ dst,
                            const float* __restrict__ dis, const float* __restrict__ h,
                            float* __restrict__ agg) {
    long long t = (long long)blockIdx.x * blockDim.x + threadIdx.x;
    long long e = t >> 5;
    int c = (int)(t & 31);
    if (e >= GCN_N_EDGES) return;
    int s = src[e];
    int d = dst[e];
    float norm = dis[s] * dis[d];
    const float4* hs = (const float4*)(h + (size_t)s * GCN_FEAT);
    float4 v = hs[c];
    float* ad = agg + (size_t)d * GCN_FEAT + (size_t)c * 4;
    atomicAdd(ad + 0, v.x * norm);
    atomicAdd(ad + 1, v.y * norm);
    atomicAdd(ad + 2, v.z * norm);
    atomicAdd(ad + 3, v.w * norm);
}

// ---------------- finalize: out = relu(agg + h*dis^2 + bias) (in-place into agg ok) ----

__global__ void gcn_finalize(const float* __restrict__ h, const float* __restrict__ agg,
                             const float* __restrict__ dis, const float* __restrict__ bias,
                             float* __restrict__ out, int n) {
    long long t = (long long)blockIdx.x * blockDim.x + threadIdx.x;
    if (t >= (long long)n * GCN_FEAT) return;
    int node = (int)(t >> 7);
    int f = (int)(t & (GCN_FEAT - 1));
    float ds = dis[node];
    float v = agg[t] + h[t] * (ds * ds) + bias[f];
    out[t] = fmaxf(v, 0.0f);
}

// ---------------- CORAL head: logits[i,j] = dot(h[i], w_fc) + th_bias[j] --------------

__global__ void gcn_head(const float* __restrict__ h, const float* __restrict__ wfc,
                         const float* __restrict__ th, float* __restrict__ out, int n) {
    int i = blockIdx.x * blockDim.x + threadIdx.x;
    if (i >= n) return;
    const float4* hr = (const float4*)(h + (size_t)i * GCN_FEAT);
    const float4* wr = (const float4*)wfc;
    float s = 0.0f;
#pragma unroll 8
    for (int f = 0; f < GCN_FEAT / 4; ++f) {
        float4 a = hr[f];
        float4 b = wr[f];
        s += a.x * b.x + a.y * b.y + a.z * b.z + a.w * b.w;
    }
#pragma unroll
    for (int j = 0; j < GCN_KOUT; ++j)
        out[(size_t)i * GCN_KOUT + j] = s + th[j];
}

// ---------------- launch ----------------

extern "C" void kernel_launch(void* const* d_in, const int* in_sizes, int n_in,
                              void* d_out, int out_size, void* d_ws, size_t ws_size,
                              hipStream_t stream) {
    const float* x    = (const float*)d_in[0];
    const int*   edge = (const int*)d_in[1];
    const float* W1   = (const float*)d_in[2];
    const float* b1   = (const float*)d_in[3];
    const float* W2   = (const float*)d_in[4];
    const float* b2   = (const float*)d_in[5];
    const float* wfc  = (const float*)d_in[6];
    const float* th   = (const float*)d_in[7];
    float* out = (float*)d_out;

    const int* src = edge;
    const int* dst = edge + GCN_N_EDGES;

    const int N = GCN_N_NODES;
    const int E = GCN_N_EDGES;
    const size_t NF = (size_t)N * GCN_FEAT;  // 12.8M floats

    // workspace layout (floats): dis[pad 102400] | h[NF] | agg[NF]  (~103 MB)
    float* dis = (float*)d_ws;
    float* h   = dis + 102400;
    float* agg = h + NF;

    const int B = 256;
    const int gN    = (N + B - 1) / B;
    const int gE    = (E + B - 1) / B;
    const int gNF   = (int)((NF + B - 1) / B);
    const long long scat_threads = (long long)E * 32;
    const int gScat = (int)((scat_threads + B - 1) / B);
    // GEMM: 8 waves/block, one 16-row x 128-col strip per wave; strips = N/16 = 6250
    const int strips = N / 16;
    const int gGemm = (strips + 7) / 8;  // 782 (tail waves exit uniformly)

    // degree -> dis = 1/sqrt(deg+1)
    gcn_zero_f32<<<gN, B, 0, stream>>>(dis, N);
    gcn_deg_count<<<gE, B, 0, stream>>>(dst, dis, E);
    gcn_deg_finalize<<<gN, B, 0, stream>>>(dis, N);

    // ---- layer 1 ----
    gcn_gemm128_wmma<<<gGemm, B, 0, stream>>>(x, W1, h, N);
    gcn_zero_f32<<<gNF, B, 0, stream>>>(agg, (int)NF);
    gcn_scatter<<<gScat, B, 0, stream>>>(src, dst, dis, h, agg);
    gcn_finalize<<<gNF, B, 0, stream>>>(h, agg, dis, b1, agg, N);  // agg = relu(act1)

    // ---- layer 2 ----
    gcn_gemm128_wmma<<<gGemm, B, 0, stream>>>(agg, W2, h, N);      // h = act1 @ W2
    gcn_zero_f32<<<gNF, B, 0, stream>>>(agg, (int)NF);
    gcn_scatter<<<gScat, B, 0, stream>>>(src, dst, dis, h, agg);
    gcn_finalize<<<gNF, B, 0, stream>>>(h, agg, dis, b2, agg, N);  // agg = relu(act2)

    // ---- CORAL head ----
    gcn_head<<<gN, B, 0, stream>>>(agg, wfc, th, out, N);

    (void)in_sizes; (void)n_in; (void)out_size; (void)ws_size;
}